// RGBAKernelConv_6313601925505
// MI455X (gfx1250) — compile-verified
//
#include <hip/hip_runtime.h>

typedef __attribute__((ext_vector_type(2))) float v2f;
typedef __attribute__((ext_vector_type(8))) float v8f;

typedef __attribute__((address_space(1))) int as1_int;
typedef __attribute__((address_space(3))) int as3_int;

#define IMG_H 512
#define IMG_W 512
#define TILE 64
#define PSTRIDE 68   // 66 halo cols, padded to 68 (conflict padding)
#define VSTRIDE 84   // 80 V cols (5 x 16-col WMMA tiles), padded

#if __has_builtin(__builtin_amdgcn_global_load_async_to_lds_b32)
#define USE_ASYNC 1
#endif

// Stage a 66x66 haloed tile (global rows h0-1..h0+64, cols w0-1..w0+64) into LDS,
// zero-filling outside the image.
__device__ __forceinline__ void stage_tile(const float* __restrict__ src,
                                           float* dst, int h0, int w0, int tid) {
  for (int idx = tid; idx < 66 * 66; idx += 256) {
    int r = idx / 66, c = idx - r * 66;
    int gh = h0 - 1 + r, gw = w0 - 1 + c;
    float* d = dst + r * PSTRIDE + c;
    if (gh >= 0 && gh < IMG_H && gw >= 0 && gw < IMG_W) {
#ifdef USE_ASYNC
      __builtin_amdgcn_global_load_async_to_lds_b32(
          (as1_int*)(src + gh * IMG_W + gw), (as3_int*)d, 0, 0);
#else
      *d = src[gh * IMG_W + gw];
#endif
    } else {
      *d = 0.0f;
    }
  }
}

__device__ __forceinline__ void drain_async() {
#ifdef USE_ASYNC
  asm volatile("s_wait_asynccnt 0" ::: "memory");
#endif
}

__global__ __launch_bounds__(256) void rgba_box_kernel(const float* __restrict__ in,
                                                       float* __restrict__ out) {
  __shared__ float sA[66 * PSTRIDE];  // alpha tile (haloed)
  __shared__ float sP[66 * PSTRIDE];  // channel tile, then product in-place
  __shared__ float sV[64 * VSTRIDE];  // vertical 3-sums

  const int tid  = threadIdx.x;
  const int lane = tid & 31;
  const int wave = tid >> 5;
  const int m    = lane & 15;   // N (col) / M (row) index within fragments
  const int half = lane >> 4;   // lane group 0 or 1

  const int w0 = blockIdx.x * TILE;
  const int h0 = blockIdx.y * TILE;
  const int b  = blockIdx.z;

  // alpha = channel 3
  stage_tile(in + ((size_t)b * 4 + 3) * IMG_H * IMG_W, sA, h0, w0, tid);
  drain_async();
  __syncthreads();

  for (int ch = 0; ch < 3; ++ch) {
    stage_tile(in + ((size_t)b * 4 + ch) * IMG_H * IMG_W, sP, h0, w0, tid);
    drain_async();
    __syncthreads();

    // P = rgb_c * alpha (in place)
    for (int idx = tid; idx < 66 * 66; idx += 256) {
      int r = idx / 66, c = idx - r * 66;
      int o = r * PSTRIDE + c;
      sP[o] = sP[o] * sA[o];
    }
    __syncthreads();

    // Vertical 3-sum via WMMA: V(16x16) = T_tri(16x16) @ P(16x16), K split into 4 slices.
    // 4 row-tiles x 5 col-tiles = 20 tiles, round-robined over 8 waves.
    for (int t = wave; t < 20; t += 8) {
      const int r0 = (t & 3) * 16;       // local output row base
      const int c0 = (t >> 2) * 16;      // V column base (V col j <-> global col w0+j-1)
      int jc = c0 + m; if (jc > 67) jc = 67;   // clamp; cols >65 are never consumed

      v8f acc = {};
      for (int k = 0; k < 4; ++k) {
        const int kl = 4 * k + 2 * half;  // K index held by .x; .y holds kl+1
        v2f a, bf;
        // A (tridiagonal ones), 16x4 layout: lane -> M=m, VGPR v -> K = 4k + v + 2*half
        int d0 = m - kl;       d0 = d0 < 0 ? -d0 : d0;
        int d1 = m - (kl + 1); d1 = d1 < 0 ? -d1 : d1;
        a.x = (d0 <= 1) ? 1.0f : 0.0f;
        a.y = (d1 <= 1) ? 1.0f : 0.0f;
        // B (data), 4x16 layout: lane -> N=m, VGPR v -> K = 4k + v + 2*half.
        // P buffer row (local K + 1) holds global row h0 + r0 + K (halo offset +1).
        bf.x = sP[(r0 + 1 + kl) * PSTRIDE + jc];
        bf.y = sP[(r0 + 2 + kl) * PSTRIDE + jc];
        acc = __builtin_amdgcn_wmma_f32_16x16x4_f32(
            /*neg_a=*/false, a, /*neg_b=*/false, bf,
            /*c_mod=*/(short)0, acc, /*reuse_a=*/false, /*reuse_b=*/false);
      }
      // Boundary fixes: out row 0 += P row above tile; out row 15 += P row below tile.
      // C layout: VGPR r holds row r (lanes 0-15) and row r+8 (lanes 16-31).
      float top = sP[(r0 + 0)  * PSTRIDE + jc];   // global row h0 + r0 - 1
      float bot = sP[(r0 + 17) * PSTRIDE + jc];   // global row h0 + r0 + 16
      acc[0] += (half == 0) ? top : 0.0f;
      acc[7] += (half == 1) ? bot : 0.0f;

      for (int r = 0; r < 8; ++r)
        sV[(r0 + r + 8 * half) * VSTRIDE + c0 + m] = acc[r];
    }
    __syncthreads();

    // Horizontal 3-sum + coalesced float4 stores. V col j <-> global col w0+j-1,
    // so out(row, w) = V[row][w] + V[row][w+1] + V[row][w+2].
    float* dst = out + (((size_t)b * 3 + ch) * IMG_H + h0) * IMG_W + w0;
    for (int i = tid; i < 64 * 16; i += 256) {
      int row = i >> 4;
      int q = (i & 15) * 4;
      const float* v = sV + row * VSTRIDE + q;
      float4 o;
      o.x = v[0] + v[1] + v[2];
      o.y = v[1] + v[2] + v[3];
      o.z = v[2] + v[3] + v[4];
      o.w = v[3] + v[4] + v[5];
      *(float4*)(dst + row * IMG_W + q) = o;
    }
    __syncthreads();  // sP/sV reused next channel
  }
}

extern "C" void kernel_launch(void* const* d_in, const int* in_sizes, int n_in,
                              void* d_out, int out_size, void* d_ws, size_t ws_size,
                              hipStream_t stream) {
  (void)in_sizes; (void)n_in; (void)out_size; (void)d_ws; (void)ws_size;
  const float* in = (const float*)d_in[0];
  float* out = (float*)d_out;
  dim3 grid(IMG_W / TILE, IMG_H / TILE, 16);  // 8 x 8 tiles x 16 batches
  dim3 block(256);
  hipLaunchKernelGGL(rgba_box_kernel, grid, block, 0, stream, in, out);
}